// ConstGCN_15582141350766
// MI455X (gfx1250) — compile-verified
//
#include <hip/hip_runtime.h>
#include <hip/hip_bf16.h>

// ConstGCN fused kernel for gfx1250 (MI455X).
//
// Key identity: softmax over axis=2 summed over axis=2 == 1, so the reference
// collapses to  out = relu(text + mean_k(emb_table[labels]) @ fc_W.T + fc_b).
// const_mat / attn_W / attn_b are mathematically irrelevant to the output.
//
// B=16, L=2048, D=256, CD=128, K=8, CN=100.
// GEMM E[32768x128] x fc_W.T[128x256] done with V_WMMA_F32_16X16X4_F32.

#define NB  16
#define NL  2048
#define ND  256
#define NCD 128
#define NCN 100
#define NK  8
#define TPAD 130            // padded emb_table row stride (floats): bank-conflict-free, 8B aligned
#define WAVES_PER_BLOCK 8
#define MTILES (NB * NL / 16)                 // 2048
#define NBLOCKS (MTILES / WAVES_PER_BLOCK)    // 256

typedef float v2f __attribute__((ext_vector_type(2)));
typedef float v8f __attribute__((ext_vector_type(8)));

__global__ __launch_bounds__(256) void constgcn_fused(
    const float* __restrict__ text,    // [B*L*D]
    const int*   __restrict__ labels,  // [B*L*K]
    const float* __restrict__ table,   // [CN*CD]
    const float* __restrict__ fcW,     // [D*CD]  (row-major == B^T in n-major form)
    const float* __restrict__ fcb,     // [D]
    float* __restrict__ out)           // [B*L*D]
{
    extern __shared__ float smem[];    // emb_table, padded: NCN * TPAD floats (52 KB)

    // ---- cooperative load of emb_table into LDS with padded stride ----
    for (int i = threadIdx.x; i < NCN * NCD; i += blockDim.x) {
        const int r = i >> 7;          // / NCD
        const int c = i & (NCD - 1);   // % NCD
        smem[r * TPAD + c] = table[i];
    }
    __syncthreads();

    const int lane = threadIdx.x & 31;
    const int wave = threadIdx.x >> 5;
    const int mtile = blockIdx.x * WAVES_PER_BLOCK + wave;  // 0..2047
    const int m0 = mtile * 16;                              // global row base
    const int Mrow = lane & 15;                             // A-matrix row for this lane
    const int h = lane >> 4;                                // K-half selector (A/B), +8 row sel (C/D)
    const int m = m0 + Mrow;

    // ---- labels for this lane's A row (two int4 loads, 16B aligned) ----
    const int4* lp = (const int4*)(labels + m * NK);
    const int4 p0 = lp[0];
    const int4 p1 = lp[1];
    int off[NK];
    off[0] = p0.x * TPAD; off[1] = p0.y * TPAD; off[2] = p0.z * TPAD; off[3] = p0.w * TPAD;
    off[4] = p1.x * TPAD; off[5] = p1.y * TPAD; off[6] = p1.z * TPAD; off[7] = p1.w * TPAD;

    // ---- gather-mean: build A fragment (16x128 fp32) directly in WMMA layout ----
    // lane covers K = {kk*4 + 2h, kk*4 + 2h + 1} for kk = 0..31
    v2f a[32];
#pragma unroll
    for (int kk = 0; kk < 32; ++kk) {
        const int kb = kk * 4 + 2 * h;
        float x = 0.0f, y = 0.0f;
#pragma unroll
        for (int t = 0; t < NK; ++t) {
            const v2f e2 = *(const v2f*)(&smem[off[t] + kb]);   // ds_load_b64
            x += e2.x;
            y += e2.y;
        }
        a[kk].x = x * 0.125f;
        a[kk].y = y * 0.125f;
    }

    // ---- GEMM: 16 N-tiles of 16 cols; K=128 -> 32 x V_WMMA_F32_16X16X4_F32 ----
    const int nl = lane & 15;          // B/C/D column within tile for this lane
#pragma unroll 1
    for (int nt = 0; nt < 16; ++nt) {
        const int n = nt * 16 + nl;
        const float* __restrict__ wrow = fcW + n * NCD;   // B^T row n (contiguous K)

        v8f c = {0.f, 0.f, 0.f, 0.f, 0.f, 0.f, 0.f, 0.f};
#pragma unroll
        for (int kk = 0; kk < 32; ++kk) {
            const v2f b = *(const v2f*)(wrow + kk * 4 + 2 * h);  // global_load_b64, L2-hot
            c = __builtin_amdgcn_wmma_f32_16x16x4_f32(
                    /*neg_a=*/false, a[kk],
                    /*neg_b=*/false, b,
                    /*c_mod=*/(short)0, c,
                    /*reuse_a=*/false, /*reuse_b=*/false);
        }

        // ---- fused epilogue: + text + bias, ReLU, store ----
        const float bias = fcb[n];
        const int colbase = nt * 16 + nl;
#pragma unroll
        for (int r = 0; r < 8; ++r) {
            const int row = m0 + r + 8 * h;            // C/D row layout
            const int idx = row * ND + colbase;
            const float v = c[r] + text[idx] + bias;
            out[idx] = fmaxf(v, 0.0f);
        }
    }
}

extern "C" void kernel_launch(void* const* d_in, const int* in_sizes, int n_in,
                              void* d_out, int out_size, void* d_ws, size_t ws_size,
                              hipStream_t stream) {
    // setup_inputs order: text, const_mat, const_labels, emb_table, attn_W, attn_b, fc_W, fc_b
    const float* text   = (const float*)d_in[0];
    // d_in[1] const_mat  : unused (softmax rows sum to 1 exactly)
    const int*   labels = (const int*)  d_in[2];
    const float* table  = (const float*)d_in[3];
    // d_in[4] attn_W, d_in[5] attn_b : unused (cancel out of the output)
    const float* fcW    = (const float*)d_in[6];
    const float* fcb    = (const float*)d_in[7];
    float* out = (float*)d_out;

    const size_t lds_bytes = (size_t)NCN * TPAD * sizeof(float);  // 52,000 B
    constgcn_fused<<<dim3(NBLOCKS), dim3(256), lds_bytes, stream>>>(
        text, labels, table, fcW, fcb, out);
}